// color_counter_43757126812179
// MI455X (gfx1250) — compile-verified
//
#include <hip/hip_runtime.h>
#include <stdint.h>
#include <math.h>

#define BINS  128
#define B3    (128 * 128 * 128)        // 2097152 bins
#define HNUM  4096
#define WNUM  4096
#define NPIX  (HNUM * WNUM)            // 16777216 pixels

// ---- workspace layout (float offsets) ----
// [0, B3)         : full histogram, later overwritten by log_ratios
// [B3, 2*B3)      : lines histogram
// [2*B3, 2*B3+8)  : scalars: s[0]=sum(lines), s[1]=avg_log_ratio
// bytes (2*B3+8)*4 ... +B3 : filter LUT (uint8 per bin)
#define OFF_FULL       0
#define OFF_LINES      B3
#define OFF_SCAL       (2 * B3)
#define OFF_FILT_BYTES ((size_t)(2 * B3 + 8) * 4)

#define TPB             256
#define WAVES_PER_BLOCK (TPB / 32)
#define HIST_BLOCKS     2048
#define TOTAL_WAVES     (HIST_BLOCKS * WAVES_PER_BLOCK)   // 16384 waves
#define PIX_PER_WAVE    128                               // per buffer
#define HIST_ITERS      (NPIX / (TOTAL_WAVES * PIX_PER_WAVE))  // 8, exact

// -------------------------------------------------------------------------
// Kernel 1: histogram build. Each wave streams 128 pixels/iteration through
// a private LDS double buffer using CDNA5 async global->LDS b128 loads
// (ASYNCcnt), then scatters agent-scope float atomics into the L2-resident
// histograms.
// -------------------------------------------------------------------------
__global__ __launch_bounds__(TPB) void hist_kernel(const int* __restrict__ img,
                                                   const float* __restrict__ mask,
                                                   float* __restrict__ hfull,
                                                   float* __restrict__ hlines) {
    // per wave: 2 buffers x (96 img int4 + 32 mask int4) = 2 x 2048 B
    __shared__ int4 smem[WAVES_PER_BLOCK][2][128];

    const int lane  = threadIdx.x & 31;
    const int wid   = threadIdx.x >> 5;
    const int gwave = blockIdx.x * WAVES_PER_BLOCK + wid;

    const char* imgB = (const char*)img;
    const char* mskB = (const char*)mask;

    auto issue = [&](int buf, int iter) {
        const uint32_t pixBase =
            (uint32_t)(gwave + iter * TOTAL_WAVES) * (uint32_t)PIX_PER_WAVE;
        const uint32_t ldsBase = (uint32_t)(uintptr_t)(&smem[wid][buf][0]);
        const uint32_t imgOff  = pixBase * 12u;  // 3 x int32 per pixel
        // 96 x 16B img chunks per wave -> 3 async b128 per lane
#pragma unroll
        for (int j = 0; j < 3; ++j) {
            uint32_t lds = ldsBase + (uint32_t)(j * 32 + lane) * 16u;
            uint32_t off = imgOff + (uint32_t)(j * 32 + lane) * 16u;
            asm volatile("global_load_async_to_lds_b128 %0, %1, %2"
                         :: "v"(lds), "v"(off), "s"(imgB) : "memory");
        }
        // 32 x 16B mask chunks -> 1 async b128 per lane
        uint32_t ldsM = ldsBase + (uint32_t)(96 + lane) * 16u;
        uint32_t offM = pixBase * 4u + (uint32_t)lane * 16u;
        asm volatile("global_load_async_to_lds_b128 %0, %1, %2"
                     :: "v"(ldsM), "v"(offM), "s"(mskB) : "memory");
    };

    issue(0, 0);
    int buf = 0;
#pragma unroll 1
    for (int iter = 0; iter < HIST_ITERS; ++iter) {
        if (iter + 1 < HIST_ITERS) {
            issue(buf ^ 1, iter + 1);
            // 4 async-load instructions outstanding for the next buffer;
            // ASYNCcnt<=4 guarantees the current buffer landed (in-order).
            asm volatile("s_wait_asynccnt 4" ::: "memory");
        } else {
            asm volatile("s_wait_asynccnt 0" ::: "memory");
        }

        const int4* wb = &smem[wid][buf][0];
        // lane processes pixels [lane*4, lane*4+4): 12 ints = 3 int4
        int4 a = wb[3 * lane + 0];
        int4 b = wb[3 * lane + 1];
        int4 c = wb[3 * lane + 2];
        float4 m = *(const float4*)(&wb[96 + lane]);

        uint32_t i0 = ((uint32_t)(a.x >> 1) << 14) | ((uint32_t)(a.y >> 1) << 7) | (uint32_t)(a.z >> 1);
        uint32_t i1 = ((uint32_t)(a.w >> 1) << 14) | ((uint32_t)(b.x >> 1) << 7) | (uint32_t)(b.y >> 1);
        uint32_t i2 = ((uint32_t)(b.z >> 1) << 14) | ((uint32_t)(b.w >> 1) << 7) | (uint32_t)(c.x >> 1);
        uint32_t i3 = ((uint32_t)(c.y >> 1) << 14) | ((uint32_t)(c.z >> 1) << 7) | (uint32_t)(c.w >> 1);

        __hip_atomic_fetch_add(&hfull[i0], 1.0f, __ATOMIC_RELAXED, __HIP_MEMORY_SCOPE_AGENT);
        __hip_atomic_fetch_add(&hfull[i1], 1.0f, __ATOMIC_RELAXED, __HIP_MEMORY_SCOPE_AGENT);
        __hip_atomic_fetch_add(&hfull[i2], 1.0f, __ATOMIC_RELAXED, __HIP_MEMORY_SCOPE_AGENT);
        __hip_atomic_fetch_add(&hfull[i3], 1.0f, __ATOMIC_RELAXED, __HIP_MEMORY_SCOPE_AGENT);
        __hip_atomic_fetch_add(&hlines[i0], m.x, __ATOMIC_RELAXED, __HIP_MEMORY_SCOPE_AGENT);
        __hip_atomic_fetch_add(&hlines[i1], m.y, __ATOMIC_RELAXED, __HIP_MEMORY_SCOPE_AGENT);
        __hip_atomic_fetch_add(&hlines[i2], m.z, __ATOMIC_RELAXED, __HIP_MEMORY_SCOPE_AGENT);
        __hip_atomic_fetch_add(&hlines[i3], m.w, __ATOMIC_RELAXED, __HIP_MEMORY_SCOPE_AGENT);

        buf ^= 1;
    }
}

// -------------------------------------------------------------------------
// Block reduction helper (256 threads)
// -------------------------------------------------------------------------
__device__ __forceinline__ float block_reduce(float v, float* sred) {
    const int t = threadIdx.x;
    sred[t] = v;
    __syncthreads();
#pragma unroll
    for (int s = TPB / 2; s > 0; s >>= 1) {
        if (t < s) sred[t] += sred[t + s];
        __syncthreads();
    }
    return sred[0];
}

// Kernel 2: sum(lines) -> scal[0]
__global__ __launch_bounds__(TPB) void sum_kernel(const float4* __restrict__ lines4,
                                                  float* __restrict__ scal) {
    __shared__ float sred[TPB];
    const int tid = blockIdx.x * TPB + threadIdx.x;   // B3/4 threads total
    float4 v = lines4[tid];
    float p = (v.x + v.y) + (v.z + v.w);
    float b = block_reduce(p, sred);
    if (threadIdx.x == 0)
        __hip_atomic_fetch_add(&scal[0], b, __ATOMIC_RELAXED, __HIP_MEMORY_SCOPE_AGENT);
}

// Kernel 3: log_ratios (overwrite full histogram in place) + weighted avg
__global__ __launch_bounds__(TPB) void logratio_kernel(float4* __restrict__ hf,      // in: full, out: log_ratios
                                                       const float4* __restrict__ hl,
                                                       float* __restrict__ scal) {
    __shared__ float sred[TPB];
    const int tid = blockIdx.x * TPB + threadIdx.x;   // B3/4 threads total

    const float Sl    = scal[0] + 1e-10f * (float)B3;            // sum(lines + 1e-10)
    const float Sf    = (float)((long long)NPIX + (long long)B3); // sum(full + 1.0), exact
    const float invSl = 1.0f / Sl;

    float4 l = hl[tid];
    float4 f = hf[tid];
    l.x += 1e-10f; l.y += 1e-10f; l.z += 1e-10f; l.w += 1e-10f;
    f.x += 1.0f;   f.y += 1.0f;   f.z += 1.0f;   f.w += 1.0f;

    float4 lr;
    lr.x = logf((l.x * Sf) / (f.x * Sl));
    lr.y = logf((l.y * Sf) / (f.y * Sl));
    lr.z = logf((l.z * Sf) / (f.z * Sl));
    lr.w = logf((l.w * Sf) / (f.w * Sl));
    hf[tid] = lr;

    float p = (l.x * lr.x + l.y * lr.y + l.z * lr.z + l.w * lr.w) * invSl;
    float b = block_reduce(p, sred);
    if (threadIdx.x == 0)
        __hip_atomic_fetch_add(&scal[1], b, __ATOMIC_RELAXED, __HIP_MEMORY_SCOPE_AGENT);
}

// Kernel 4: bake threshold into a 2 MB byte LUT (4x denser than f32 gathers)
__global__ __launch_bounds__(TPB) void filter_kernel(const float4* __restrict__ lr4,
                                                     const float* __restrict__ scal,
                                                     uchar4* __restrict__ filt4) {
    const int tid = blockIdx.x * TPB + threadIdx.x;   // B3/4 threads total
    const float avg = scal[1];
    float4 v = lr4[tid];
    uchar4 o;
    o.x = (unsigned char)(v.x > avg ? 1 : 0);
    o.y = (unsigned char)(v.y > avg ? 1 : 0);
    o.z = (unsigned char)(v.z > avg ? 1 : 0);
    o.w = (unsigned char)(v.w > avg ? 1 : 0);
    filt4[tid] = o;
}

// Kernel 5: per-pixel gather of the boolean filter -> float output
__global__ __launch_bounds__(TPB) void gather_kernel(const int4* __restrict__ img4,
                                                     const uint8_t* __restrict__ filt,
                                                     float4* __restrict__ out4) {
    const int t = blockIdx.x * TPB + threadIdx.x;     // NPIX/4 threads total
    int4 a = img4[3 * t + 0];
    int4 b = img4[3 * t + 1];
    int4 c = img4[3 * t + 2];

    uint32_t i0 = ((uint32_t)(a.x >> 1) << 14) | ((uint32_t)(a.y >> 1) << 7) | (uint32_t)(a.z >> 1);
    uint32_t i1 = ((uint32_t)(a.w >> 1) << 14) | ((uint32_t)(b.x >> 1) << 7) | (uint32_t)(b.y >> 1);
    uint32_t i2 = ((uint32_t)(b.z >> 1) << 14) | ((uint32_t)(b.w >> 1) << 7) | (uint32_t)(c.x >> 1);
    uint32_t i3 = ((uint32_t)(c.y >> 1) << 14) | ((uint32_t)(c.z >> 1) << 7) | (uint32_t)(c.w >> 1);

    float4 o;
    o.x = filt[i0] ? 1.0f : 0.0f;
    o.y = filt[i1] ? 1.0f : 0.0f;
    o.z = filt[i2] ? 1.0f : 0.0f;
    o.w = filt[i3] ? 1.0f : 0.0f;
    out4[t] = o;
}

// -------------------------------------------------------------------------
extern "C" void kernel_launch(void* const* d_in, const int* in_sizes, int n_in,
                              void* d_out, int out_size, void* d_ws, size_t ws_size,
                              hipStream_t stream) {
    const int*   img  = (const int*)d_in[0];
    const float* mask = (const float*)d_in[1];
    // d_in[2]/d_in[3] (zero-initialized full/lines) are replaced by workspace
    float*   ws     = (float*)d_ws;
    float*   hfull  = ws + OFF_FULL;    // later: log_ratios
    float*   hlines = ws + OFF_LINES;
    float*   scal   = ws + OFF_SCAL;
    uint8_t* filt   = (uint8_t*)d_ws + OFF_FILT_BYTES;

    // zero histograms + scalars (graph-capturable)
    hipMemsetAsync(d_ws, 0, (size_t)(2 * B3 + 8) * sizeof(float), stream);

    hist_kernel<<<HIST_BLOCKS, TPB, 0, stream>>>(img, mask, hfull, hlines);

    const int binBlocks = B3 / 4 / TPB;   // 2048
    sum_kernel<<<binBlocks, TPB, 0, stream>>>((const float4*)hlines, scal);
    logratio_kernel<<<binBlocks, TPB, 0, stream>>>((float4*)hfull, (const float4*)hlines, scal);
    filter_kernel<<<binBlocks, TPB, 0, stream>>>((const float4*)hfull, scal, (uchar4*)filt);

    const int pixBlocks = NPIX / 4 / TPB; // 16384
    gather_kernel<<<pixBlocks, TPB, 0, stream>>>((const int4*)img, filt, (float4*)d_out);
}